// GraphSAGEBackbone_62783831933158
// MI455X (gfx1250) — compile-verified
//
#include <hip/hip_runtime.h>
#include <stdint.h>

typedef __attribute__((ext_vector_type(2))) float v2f;
typedef __attribute__((ext_vector_type(4))) float v4f;
typedef __attribute__((ext_vector_type(8))) float v8f;
typedef __attribute__((ext_vector_type(4))) int   v4i;

#define D 128  // D_IN == D_HID == D_OUT == 128

#if defined(__has_builtin)
#if __has_builtin(__builtin_amdgcn_global_load_async_to_lds_b128) && \
    __has_builtin(__builtin_amdgcn_s_wait_asynccnt)
#define USE_ASYNC_LDS 1
#endif
#endif

// ---------------- in-degree count ----------------
__global__ void k_count(const int* __restrict__ dst, float* __restrict__ cnt, int E) {
  int e = blockIdx.x * blockDim.x + threadIdx.x;
  if (e < E) atomicAdd(cnt + dst[e], 1.0f);
}

// ---------------- feature scatter-add: one wave per edge ----------------
__global__ void k_scatter(const float* __restrict__ X, const int* __restrict__ src,
                          const int* __restrict__ dst, float* __restrict__ summed, int E) {
  int t = blockIdx.x * blockDim.x + threadIdx.x;
  int e = t >> 5;
  int lane = t & 31;
  if (e >= E) return;
  long long s = src[e];
  long long d = dst[e];
  const float* xp = X + s * D + lane * 4;
  float* op = summed + d * D + lane * 4;
  v4f v = *(const v4f*)xp;                // coalesced 512B per wave
  atomicAdd(op + 0, v.x);                 // non-returning global_atomic_add_f32
  atomicAdd(op + 1, v.y);
  atomicAdd(op + 2, v.z);
  atomicAdd(op + 3, v.w);
}

// ---------------- weight pre-pack: Wpk[kp*128+n] = {W[2kp][n], W[2kp+1][n]} ----------------
// Matches the 16x16x4 f32 B-fragment layout (k0+2*kh is always even), so the
// GEMM's LDS staging becomes a straight block copy (async-LDS friendly).
__global__ void k_pack(const float* __restrict__ W, v2f* __restrict__ Wpk) {
  int i = blockIdx.x * blockDim.x + threadIdx.x;
  if (i < 64 * D) {
    int kp = i >> 7;
    int n  = i & (D - 1);
    v2f p;
    p.x = W[(2 * kp) * D + n];
    p.y = W[(2 * kp + 1) * D + n];
    Wpk[i] = p;
  }
}

// Stage one packed 64KB weight matrix into LDS.
static __device__ __forceinline__ void stage_weights(const v2f* __restrict__ Wpk,
                                                     v2f* lds, int tid) {
#if defined(USE_ASYNC_LDS)
  v4i* g = (v4i*)(uintptr_t)Wpk;    // 4096 x 16B global source
  v4i* l = (v4i*)lds;               // LDS destination
#pragma unroll
  for (int i = 0; i < 16; ++i) {
    int idx = tid + i * 256;
    __builtin_amdgcn_global_load_async_to_lds_b128(g + idx, l + idx, 0, 0);
  }
  __builtin_amdgcn_s_wait_asynccnt(0);   // own wave's async copies done
#else
  for (int i = tid; i < 64 * D; i += 256) lds[i] = Wpk[i];
#endif
}

// ---------------- fused SAGE GEMM: out = relu?((summed/cnt)@Wl + b + X@Wr) ----------------
// Block = 256 threads = 8 waves; each wave: 16 rows x all 128 cols (8 WMMA col-tiles).
__global__ __launch_bounds__(256)
void k_sage_gemm(const float* __restrict__ summed, const float* __restrict__ cnt,
                 const float* __restrict__ X,
                 const v2f* __restrict__ Wlpk, const float* __restrict__ bias,
                 const v2f* __restrict__ Wrpk,
                 float* __restrict__ out, int N, int do_relu) {
  __shared__ v2f ldsW[64 * D];  // 64 KB

  const int tid  = threadIdx.x;
  const int wave = tid >> 5;
  const int lane = tid & 31;
  const int nCol = lane & 15;   // A-frag row-in-tile AND B/D column-in-tile
  const int kh   = lane >> 4;   // K-half selector (lanes 16-31 hold K=2,3)

  const long long rowA  = (long long)blockIdx.x * 128 + wave * 16 + nCol;
  const long long rowAc = rowA < (long long)N ? rowA : (long long)(N - 1);

  const float c   = cnt[rowAc];
  const float inv = 1.0f / fmaxf(c, 1.0f);
  const float* __restrict__ Srow = summed + rowAc * D;
  const float* __restrict__ Xrow = X + rowAc * D;

  v8f acc[8];
#pragma unroll
  for (int t = 0; t < 8; ++t) {
    float b = bias[t * 16 + nCol];  // D-col of every acc element in this lane
    v8f bb = {b, b, b, b, b, b, b, b};
    acc[t] = bb;
  }

  // ===== pass 1: (summed / cnt) @ Wl =====
  stage_weights(Wlpk, ldsW, tid);
  __syncthreads();
#pragma unroll 4
  for (int k0 = 0; k0 < D; k0 += 4) {
    v2f a = *(const v2f*)(Srow + k0 + 2 * kh);
    a.x *= inv;
    a.y *= inv;
    const int bi = (k0 >> 1) + kh;
#pragma unroll
    for (int t = 0; t < 8; ++t) {
      v2f b = ldsW[bi * D + t * 16 + nCol];
      acc[t] = __builtin_amdgcn_wmma_f32_16x16x4_f32(false, a, false, b, (short)0,
                                                     acc[t], false, false);
    }
  }
  __syncthreads();

  // ===== pass 2: X @ Wr =====
  stage_weights(Wrpk, ldsW, tid);
  __syncthreads();
#pragma unroll 4
  for (int k0 = 0; k0 < D; k0 += 4) {
    v2f a = *(const v2f*)(Xrow + k0 + 2 * kh);
    const int bi = (k0 >> 1) + kh;
#pragma unroll
    for (int t = 0; t < 8; ++t) {
      v2f b = ldsW[bi * D + t * 16 + nCol];
      acc[t] = __builtin_amdgcn_wmma_f32_16x16x4_f32(false, a, false, b, (short)0,
                                                     acc[t], false, false);
    }
  }

  // ===== store: lane holds D rows (8*kh + r), col = t*16 + nCol =====
  const long long mBase = (long long)blockIdx.x * 128 + wave * 16 + 8 * kh;
#pragma unroll
  for (int r = 0; r < 8; ++r) {
    long long orow = mBase + r;
    if (orow < (long long)N) {
#pragma unroll
      for (int t = 0; t < 8; ++t) {
        float v = acc[t][r];
        if (do_relu) v = fmaxf(v, 0.0f);
        out[orow * D + t * 16 + nCol] = v;
      }
    }
  }
}

extern "C" void kernel_launch(void* const* d_in, const int* in_sizes, int n_in,
                              void* d_out, int out_size, void* d_ws, size_t ws_size,
                              hipStream_t stream) {
  const float* x   = (const float*)d_in[0];
  const int*   ei  = (const int*)d_in[1];
  const float* W1l = (const float*)d_in[2];
  const float* b1  = (const float*)d_in[3];
  const float* W1r = (const float*)d_in[4];
  const float* W2l = (const float*)d_in[5];
  const float* b2  = (const float*)d_in[6];
  const float* W2r = (const float*)d_in[7];

  const int N = in_sizes[0] / D;
  const int E = in_sizes[1] / 2;
  const int* src = ei;
  const int* dst = ei + E;

  float* cnt    = (float*)d_ws;
  float* summed = cnt + N;
  float* h      = summed + (size_t)N * D;
  v2f*   wpk    = (v2f*)(h + (size_t)N * D);  // 4 x 8192 v2f (4 x 64KB)
  v2f* W1lpk = wpk;
  v2f* W1rpk = wpk + 1 * 64 * D;
  v2f* W2lpk = wpk + 2 * 64 * D;
  v2f* W2rpk = wpk + 3 * 64 * D;
  float* outp = (float*)d_out;

  // zero degree counts + scatter accumulator (contiguous in ws)
  (void)hipMemsetAsync(cnt, 0, ((size_t)N + (size_t)N * D) * sizeof(float), stream);

  // pre-pack weights into WMMA B-fragment-friendly pair-interleaved layout
  const int pb = (64 * D + 255) / 256;
  k_pack<<<pb, 256, 0, stream>>>(W1l, W1lpk);
  k_pack<<<pb, 256, 0, stream>>>(W1r, W1rpk);
  k_pack<<<pb, 256, 0, stream>>>(W2l, W2lpk);
  k_pack<<<pb, 256, 0, stream>>>(W2r, W2rpk);

  const int gb = (N + 127) / 128;

  // ---- layer 1 ----
  k_count<<<(E + 255) / 256, 256, 0, stream>>>(dst, cnt, E);
  k_scatter<<<(E * 32 + 255) / 256, 256, 0, stream>>>(x, src, dst, summed, E);
  k_sage_gemm<<<gb, 256, 0, stream>>>(summed, cnt, x, W1lpk, b1, W1rpk, h, N, 1);

  // ---- layer 2 (degree counts unchanged) ----
  (void)hipMemsetAsync(summed, 0, (size_t)N * D * sizeof(float), stream);
  k_scatter<<<(E * 32 + 255) / 256, 256, 0, stream>>>(h, src, dst, summed, E);
  k_sage_gemm<<<gb, 256, 0, stream>>>(summed, cnt, h, W2lpk, b2, W2rpk, outp, N, 0);
}